// SimpleGrid_9646496547661
// MI455X (gfx1250) — compile-verified
//
#include <hip/hip_runtime.h>

// Trilinear interpolation gather kernel for MI455X (gfx1250, wave32).
// Grid: [512,512,256,2] fp32 (512 MB). Points: [N,3] fp32. Out: sigma[N] ++ alpha[N].
// Gather-bandwidth workload: WMMA/TDM structurally inapplicable.
// Round-2 change: fuse each (z, z+1) float2 pair into ONE 16-byte load at
// zb = min(iz, SZ-2)  (8 gathers -> 4), with a select to fix the fz==0
// upper-edge clamp case. Same HBM cachelines touched, half the vmem ops.

static constexpr int SX = 512, SY = 512, SZ = 256;

// Load 16 contiguous bytes (grid[off/2] and grid[off/2+1], off in float units,
// 8-byte aligned) and z-lerp them. 'zc' = true only when iz==SZ-1 (fz==0),
// in which case the correct value is the HIGH float2 of the pair.
__device__ __forceinline__ void zlerp_pair(const float* __restrict__ g, int off,
                                           bool zc, float fz,
                                           float& cx, float& cy) {
    float t[4];
    __builtin_memcpy(t, g + off, 16);   // align 8 -> global_load_b128 (unaligned mode)
    const float lox = zc ? t[2] : t[0];
    const float loy = zc ? t[3] : t[1];
    cx = fmaf(fz, t[2] - lox, lox);
    cy = fmaf(fz, t[3] - loy, loy);
}

__global__ __launch_bounds__(256) void trilerp_kernel(
    const float*  __restrict__ xs,     // [n,3]
    const float*  __restrict__ grid,   // [512,512,256,2] floats
    const float*  __restrict__ lower,  // [3]
    const float*  __restrict__ res,    // [3]
    float*        __restrict__ sigma,  // [n]
    float*        __restrict__ alpha,  // [n]
    int n)
{
    const int i = blockIdx.x * 256 + threadIdx.x;
    if (i >= n) return;

    // Uniform scalars (s_load): same for every lane.
    const float lx = lower[0], ly = lower[1], lz = lower[2];
    const float rx = res[0],   ry = res[1],   rz = res[2];

    // Streaming position read: non-temporal (protect L2 for the grid).
    const size_t base = (size_t)i * 3u;
    const float px = __builtin_nontemporal_load(xs + base + 0);
    const float py = __builtin_nontemporal_load(xs + base + 1);
    const float pz = __builtin_nontemporal_load(xs + base + 2);

    const float gx = (px - lx) * rx;
    const float gy = (py - ly) * ry;
    const float gz = (pz - lz) * rz;

    float s = 0.0f, a = 0.0f;

    const bool valid =
        (gx >= 0.0f) && (gx <= (float)(SX - 1)) &&
        (gy >= 0.0f) && (gy <= (float)(SY - 1)) &&
        (gz >= 0.0f) && (gz <= (float)(SZ - 1));

    if (valid) {
        const float fx0 = floorf(gx), fy0 = floorf(gy), fz0 = floorf(gz);
        const float fx = gx - fx0, fy = gy - fy0, fz = gz - fz0;
        const int ix = (int)fx0, iy = (int)fy0, iz = (int)fz0;
        const int ix1 = (ix + 1 < SX) ? ix + 1 : SX - 1;
        const int iy1 = (iy + 1 < SY) ? iy + 1 : SY - 1;

        // z pair base: always load [zb, zb+1]; zc flags the fz==0 clamp case.
        const bool zc = (iz >= SZ - 1);
        const int  zb = zc ? (SZ - 2) : iz;

        // Offsets in float units (max 2^27, fits 32-bit).
        const int bx0 = ix  * (SY * SZ * 2);
        const int bx1 = ix1 * (SY * SZ * 2);
        const int by0 = iy  * (SZ * 2);
        const int by1 = iy1 * (SZ * 2);
        const int zb2 = zb * 2;

        // 4 x 16-byte gathers, back-to-back for max MLP, each z-lerped.
        float c00x, c00y, c01x, c01y, c10x, c10y, c11x, c11y;
        zlerp_pair(grid, bx0 + by0 + zb2, zc, fz, c00x, c00y);
        zlerp_pair(grid, bx0 + by1 + zb2, zc, fz, c01x, c01y);
        zlerp_pair(grid, bx1 + by0 + zb2, zc, fz, c10x, c10y);
        zlerp_pair(grid, bx1 + by1 + zb2, zc, fz, c11x, c11y);

        // lerp y, then x.
        const float c0x = fmaf(fy, c01x - c00x, c00x);
        const float c0y = fmaf(fy, c01y - c00y, c00y);
        const float c1x = fmaf(fy, c11x - c10x, c10x);
        const float c1y = fmaf(fy, c11y - c10y, c10y);

        s = fmaf(fx, c1x - c0x, c0x);
        a = fmaf(fx, c1y - c0y, c0y);
    }

    // Streaming result writes: non-temporal, coalesced per wave.
    __builtin_nontemporal_store(s, sigma + i);
    __builtin_nontemporal_store(a, alpha + i);
}

extern "C" void kernel_launch(void* const* d_in, const int* in_sizes, int n_in,
                              void* d_out, int out_size, void* d_ws, size_t ws_size,
                              hipStream_t stream) {
    (void)n_in; (void)d_ws; (void)ws_size; (void)out_size;

    const float* xs    = (const float*)d_in[0];
    const float* grid  = (const float*)d_in[1];
    const float* lower = (const float*)d_in[2];
    const float* res   = (const float*)d_in[3];

    const int n = in_sizes[0] / 3;          // 2,000,000 points
    float* sigma = (float*)d_out;           // first n outputs
    float* alpha = sigma + n;               // second n outputs

    const int block = 256;                  // 8 wave32s per block
    const int gridsz = (n + block - 1) / block;
    trilerp_kernel<<<gridsz, block, 0, stream>>>(xs, grid, lower, res, sigma, alpha, n);
}